// PagedAttention_28260884808065
// MI455X (gfx1250) — compile-verified
//
#include <hip/hip_runtime.h>

typedef float v2f __attribute__((ext_vector_type(2)));
typedef float v8f __attribute__((ext_vector_type(8)));

#define NUM_SEQS 64
#define NUM_HEADS 32
#define NUM_KV_HEADS 8
#define HEAD_SIZE 128
#define BLOCK_SIZE 16
#define MAX_BLOCKS_PER_SEQ 64
#define Q_PER_KV 4
#define ATT_SCALE 0.08838834764831845f
#define WAVES 4

// One workgroup per (kv_head, seq). 4 waves flash-decode the context pages.
// Per-wave online softmax; raw (m, l, acc) combined across waves via LDS.
// Both GEMMs use V_WMMA_F32_16X16X4_F32 (fp32 data -> exact-precision path;
// kernel is HBM-bound: ~270MB avg / 23.3TB/s ~= 12us floor).
// KV stream is read-once and larger than the 192MB L2 -> non-temporal loads;
// next page's K/V tiles are prefetched (global_prefetch_b8) while the current
// page's WMMAs run, hiding the block-table -> load dependency.
__global__ __launch_bounds__(128) void paged_attn_wmma_kernel(
    const float* __restrict__ query,
    const float* __restrict__ key_cache,     // [4096, 8, 16, 16, 8] f32
    const float* __restrict__ value_cache,   // [4096, 8, 128, 16] f32
    const int*   __restrict__ block_tables,  // [64, 64]
    const int*   __restrict__ context_lens,  // [64]
    float* __restrict__ out)                 // [64, 32, 128]
{
    const int kvh  = blockIdx.x;
    const int seq  = blockIdx.y;
    const int tid  = threadIdx.x;
    const int wave = tid >> 5;
    const int lane = tid & 31;
    const int half = lane >> 4;      // lane half selects K-rows / M-rows 8..15
    const int col  = lane & 15;      // N column (q-heads replicated x4)
    const int q    = col & 3;        // real q-head within the GQA group

    __shared__ float lds_P[WAVES][16 * 18];    // P tile, stride-18 rows (bank-safe)
    __shared__ float lds_m[WAVES][16];
    __shared__ float lds_l[WAVES][16];
    __shared__ float lds_out[WAVES][16 * 132]; // raw acc per wave, stride-132 rows

    const int  ctx     = context_lens[seq];
    const int  nblocks = (ctx + BLOCK_SIZE - 1) / BLOCK_SIZE;
    const int* btab    = block_tables + seq * MAX_BLOCKS_PER_SEQ;

    // ---- Preload Q as WMMA B-fragments (4 dims x 16 cols per step) --------
    // B layout assumption (mirrors documented A layout): vgpr0 = k rows {0,2},
    // vgpr1 = k rows {1,3}; lanes 0-15 -> k base 0, lanes 16-31 -> k base 2.
    const float* qb = query + ((size_t)seq * NUM_HEADS + kvh * Q_PER_KV + q) * HEAD_SIZE
                      + half * 2;
    v2f qfrag[32];
#pragma unroll
    for (int s = 0; s < 32; ++s)
        qfrag[s] = *(const v2f*)(qb + 4 * s);

    // key_cache per (page, kvh): [d_hi=16][t=16][d_lo=8]; element offset for
    // wmma step s (dims 4s..4s+3): (s>>1)*128 + (s&1)*4 + t*8 + (half?2:0)
    const int lane_koff = col * 8 + half * 2;
    // value_cache per (page, kvh): [d=128][t=16]; A-fragment (V^T 16d x 4t):
    const int lane_voff = col * 16 + half * 2;
    const int pf_off    = lane * 32;           // 128B/lane: 32 lanes = 4KB

    float m_run = -1e30f;
    float l_run = 0.0f;
    v8f acc[8];
#pragma unroll
    for (int dc = 0; dc < 8; ++dc)
        acc[dc] = (v8f){0.f, 0.f, 0.f, 0.f, 0.f, 0.f, 0.f, 0.f};

    for (int blk = wave; blk < nblocks; blk += WAVES) {
        const int bid = btab[blk];
        const float* kb_base = key_cache   + ((size_t)bid * NUM_KV_HEADS + kvh) * 2048;
        const float* vb_base = value_cache + ((size_t)bid * NUM_KV_HEADS + kvh) * 2048;

        // ---- prefetch next page's K/V tiles (2x 8KB) while we compute -----
        if (blk + WAVES < nblocks) {
            const int nbid = btab[blk + WAVES];
            const float* nk = key_cache   + ((size_t)nbid * NUM_KV_HEADS + kvh) * 2048;
            const float* nv = value_cache + ((size_t)nbid * NUM_KV_HEADS + kvh) * 2048;
            __builtin_prefetch(nk + pf_off, 0, 3);
            __builtin_prefetch(nk + 1024 + pf_off, 0, 3);
            __builtin_prefetch(nv + pf_off, 0, 3);
            __builtin_prefetch(nv + 1024 + pf_off, 0, 3);
        }

        // ---------------- QK^T: D[t, col] over this 16-token page ----------
        const float* kb = kb_base + lane_koff;
        v8f d = (v8f){0.f, 0.f, 0.f, 0.f, 0.f, 0.f, 0.f, 0.f};
#pragma unroll
        for (int s = 0; s < 32; ++s) {
            v2f a = __builtin_nontemporal_load(
                        (const v2f*)(kb + (s >> 1) * 128 + (s & 1) * 4));
            d = __builtin_amdgcn_wmma_f32_16x16x4_f32(
                    false, a, false, qfrag[s], (short)0, d, false, false);
        }

        // ---------------- online softmax over the page ---------------------
        float sc[8];
        float mb = -1e30f;
        const int tbase = blk * BLOCK_SIZE + half * 8;
#pragma unroll
        for (int i = 0; i < 8; ++i) {
            float si = (tbase + i < ctx) ? d[i] * ATT_SCALE : -1e30f;
            sc[i] = si;
            mb = fmaxf(mb, si);
        }
        mb = fmaxf(mb, __shfl_xor(mb, 16, 32));           // join lane halves
        const float m_new = fmaxf(m_run, mb);
        const float corr  = __expf(m_run - m_new);
        float psum = 0.0f;
#pragma unroll
        for (int i = 0; i < 8; ++i) {
            float p = __expf(sc[i] - m_new);
            psum += p;
            lds_P[wave][(i + half * 8) * 18 + col] = p;   // C-layout -> LDS
        }
        psum += __shfl_xor(psum, 16, 32);
        l_run = l_run * corr + psum;
        m_run = m_new;
#pragma unroll
        for (int dc = 0; dc < 8; ++dc)
#pragma unroll
            for (int i = 0; i < 8; ++i)
                acc[dc][i] *= corr;

        // ---------------- P @ V: acc[d, col] += V^T x P ---------------------
        const float* vb = vb_base + lane_voff;
#pragma unroll
        for (int j = 0; j < 4; ++j) {                     // 4 tokens per step
            const int k0 = 4 * j + half * 2;
            v2f pb;
            pb.x = lds_P[wave][k0 * 18 + col];            // B-layout from LDS
            pb.y = lds_P[wave][(k0 + 1) * 18 + col];
#pragma unroll
            for (int dc = 0; dc < 8; ++dc) {              // 8 chunks of 16 dims
                v2f va = __builtin_nontemporal_load(
                             (const v2f*)(vb + dc * 256 + 4 * j));
                acc[dc] = __builtin_amdgcn_wmma_f32_16x16x4_f32(
                              false, va, false, pb, (short)0, acc[dc], false, false);
            }
        }
    }

    // ---- publish raw per-wave state, then combine after one barrier -------
    if (lane < 16) {
        lds_m[wave][lane] = m_run;
        lds_l[wave][lane] = l_run;
    }
#pragma unroll
    for (int dc = 0; dc < 8; ++dc)
#pragma unroll
        for (int i = 0; i < 8; ++i)
            lds_out[wave][col * 132 + dc * 16 + half * 8 + i] = acc[dc][i];
    __syncthreads();

    float* o = out + ((size_t)seq * NUM_HEADS + kvh * Q_PER_KV) * HEAD_SIZE;
#pragma unroll
    for (int r = 0; r < 4; ++r) {
        const int flat = r * 128 + tid;    // 0..511 -> (q-head, dim)
        const int oq = flat >> 7;
        const int od = flat & 127;
        float M = -1e30f;
#pragma unroll
        for (int w = 0; w < WAVES; ++w)
            M = fmaxf(M, lds_m[w][oq]);
        float Ls = 0.0f, val = 0.0f;
#pragma unroll
        for (int w = 0; w < WAVES; ++w) {
            const float f = __expf(lds_m[w][oq] - M);
            Ls  += lds_l[w][oq] * f;
            val += lds_out[w][oq * 132 + od] * f;
        }
        o[oq * HEAD_SIZE + od] = val / Ls;
    }
}

extern "C" void kernel_launch(void* const* d_in, const int* in_sizes, int n_in,
                              void* d_out, int out_size, void* d_ws, size_t ws_size,
                              hipStream_t stream) {
    const float* query        = (const float*)d_in[0];
    const float* key_cache    = (const float*)d_in[1];
    const float* value_cache  = (const float*)d_in[2];
    const int*   block_tables = (const int*)d_in[3];
    const int*   context_lens = (const int*)d_in[4];
    float* out = (float*)d_out;

    dim3 grid(NUM_KV_HEADS, NUM_SEQS);   // 512 workgroups x 4 waves
    paged_attn_wmma_kernel<<<grid, 128, 0, stream>>>(
        query, key_cache, value_cache, block_tables, context_lens, out);
}